// Pre_EnvMap_BRDF_renderloss_pkl_15144054686503
// MI455X (gfx1250) — compile-verified
//
#include <hip/hip_runtime.h>

// ---------------------------------------------------------------------------
// EnvMap BRDF render loss for MI455X (gfx1250, wave32).
//
// Memory-bound gather kernel (~28 MB traffic, ~50 MFLOP): one thread per
// pixel, non-temporal coalesced reads of the streamed brdf/idx planes,
// 8 independent b96 gathers per (i,n) term for MLP, exact f32 accumulation.
// Loss reduction uses V_WMMA_F32_16X16X4_F32 as an exact f32 wave-reduction
// primitive (A = lane values, B = ones -> row sums), finished with one
// shfl_xor(16). Deterministic: fixed reduction order, no float atomics.
// ---------------------------------------------------------------------------

typedef __attribute__((ext_vector_type(2))) float v2f;
typedef __attribute__((ext_vector_type(8))) float v8f;

#define BB    4
#define HEE   256
#define WEE   512
#define NNN   32
#define SSS   128
#define NPIX  (SSS * SSS)          // 16384 pixels
#define DENOM (8192.0f * 4.0f * 3.0f)   // PIXELS * B * 3 = 98304

// Exact f32 sum of all 32 lane values via one f32 WMMA.
// A(16x4): lane L<16 -> A[L,0]=v, A[L,1]=0 ; lane L>=16 -> A[L-16,2]=v, A[L-16,3]=0
// B(4x16) = ones  =>  D[m,n] = v_m + v_{m+16} (rows identical across columns).
// Sum of the 8 D VGPRs: lanes 0-15 hold sum(m=0..7), lanes 16-31 hold sum(m=8..15).
__device__ __forceinline__ float wave_sum_wmma(float v) {
  v2f a; a[0] = v;    a[1] = 0.0f;
  v2f b; b[0] = 1.0f; b[1] = 1.0f;
  v8f c = {};
  c = __builtin_amdgcn_wmma_f32_16x16x4_f32(false, a, false, b,
                                            (short)0, c, false, false);
  float h = ((c[0] + c[1]) + (c[2] + c[3])) + ((c[4] + c[5]) + (c[6] + c[7]));
  h += __shfl_xor(h, 16, 32);
  return h;  // total in every lane
}

__global__ __launch_bounds__(256)
void envmap_render_main(const float* __restrict__ x, const float* __restrict__ y,
                        const float* __restrict__ bs0, const float* __restrict__ bd0,
                        const int*   __restrict__ ix0, const int*   __restrict__ iy0,
                        const float* __restrict__ w0,
                        const float* __restrict__ bs1, const float* __restrict__ bd1,
                        const int*   __restrict__ ix1, const int*   __restrict__ iy1,
                        const float* __restrict__ w1,
                        float* __restrict__ out,          // [0]=loss, [1..]=pred0, gt0
                        float* __restrict__ blockPartials) // 64 floats in d_ws
{
  __shared__ float wsh[2 * NNN];   // 64 combined weights
  __shared__ float wavePart[8];    // one partial per wave (8 waves / block)

  const int t = threadIdx.x;
  if (t < 2 * NNN) wsh[t] = (t < NNN) ? w0[t] : w1[t - NNN];
  __syncthreads();

  const int pixel = blockIdx.x * 256 + t;   // grid covers exactly NPIX

  float accP[12], accG[12];
#pragma unroll
  for (int j = 0; j < 12; ++j) { accP[j] = 0.0f; accG[j] = 0.0f; }

#pragma unroll
  for (int pass = 0; pass < 2; ++pass) {
    const float* bs = pass ? bs1 : bs0;
    const float* bd = pass ? bd1 : bd0;
    const int*   ix = pass ? ix1 : ix0;
    const int*   iy = pass ? iy1 : iy0;

    for (int n = 0; n < NNN; ++n) {
      // Streamed exactly once -> non-temporal, keep L2 for the env maps.
      const float spec = __builtin_nontemporal_load(bs + (size_t)n * NPIX + pixel);
      const float diff = __builtin_nontemporal_load(bd + (size_t)n * NPIX + pixel);
      const int   xi   = __builtin_nontemporal_load(ix + (size_t)n * NPIX + pixel);
      const int   yi   = __builtin_nontemporal_load(iy + (size_t)n * NPIX + pixel);

      const float coef = fmaf(10.0f, spec, diff) * wsh[pass * NNN + n];
      const size_t base = ((size_t)yi * WEE + (size_t)xi) * 3;

#pragma unroll
      for (int b = 0; b < BB; ++b) {
        const size_t off = (size_t)b * ((size_t)HEE * WEE * 3) + base;
        const float* px = x + off;
        const float* py = y + off;
#pragma unroll
        for (int c = 0; c < 3; ++c) {
          accP[b * 3 + c] = fmaf(coef, px[c], accP[b * 3 + c]);
          accG[b * 3 + c] = fmaf(coef, py[c], accG[b * 3 + c]);
        }
      }
    }
  }

  // pred[0] (3,128,128) then gt[0] (3,128,128), c-major, pixel row-major.
#pragma unroll
  for (int c = 0; c < 3; ++c) {
    out[1 + (size_t)c * NPIX + pixel]            = accP[c];
    out[1 + 3 * (size_t)NPIX + (size_t)c * NPIX + pixel] = accG[c];
  }

  // Squared error over all (b,c) of this pixel.
  float sq = 0.0f;
#pragma unroll
  for (int j = 0; j < 12; ++j) {
    const float d = accP[j] - accG[j];
    sq = fmaf(d, d, sq);
  }

  // Wave32 reduction via f32 WMMA (EXEC all-1s: no divergence above).
  const float wsum = wave_sum_wmma(sq);
  const int wid = t >> 5;
  if ((t & 31) == 0) wavePart[wid] = wsum;
  __syncthreads();

  if (t == 0) {
    float s = 0.0f;
#pragma unroll
    for (int w = 0; w < 8; ++w) s += wavePart[w];   // fixed order -> deterministic
    blockPartials[blockIdx.x] = s;
  }
}

// One wave reduces the 64 block partials: lane L loads p[2L], p[2L+1] into
// A(16x4) (covers all 64 values exactly), B = ones -> WMMA row sums, finish
// with shfl_xor(16).
__global__ __launch_bounds__(32)
void envmap_finalize_loss(const float* __restrict__ parts, float* __restrict__ out)
{
  const int L = threadIdx.x;
  v2f a; a[0] = parts[2 * L]; a[1] = parts[2 * L + 1];
  v2f b; b[0] = 1.0f;         b[1] = 1.0f;
  v8f c = {};
  c = __builtin_amdgcn_wmma_f32_16x16x4_f32(false, a, false, b,
                                            (short)0, c, false, false);
  float h = ((c[0] + c[1]) + (c[2] + c[3])) + ((c[4] + c[5]) + (c[6] + c[7]));
  h += __shfl_xor(h, 16, 32);
  if (L == 0) out[0] = h * (1.0f / DENOM);
}

extern "C" void kernel_launch(void* const* d_in, const int* in_sizes, int n_in,
                              void* d_out, int out_size, void* d_ws, size_t ws_size,
                              hipStream_t stream) {
  (void)in_sizes; (void)n_in; (void)out_size; (void)ws_size;

  const float* x   = (const float*)d_in[0];
  const float* y   = (const float*)d_in[1];
  const float* bs0 = (const float*)d_in[2];
  const float* bd0 = (const float*)d_in[3];
  const int*   ix0 = (const int*)  d_in[4];
  const int*   iy0 = (const int*)  d_in[5];
  const float* w0  = (const float*)d_in[6];
  const float* bs1 = (const float*)d_in[7];
  const float* bd1 = (const float*)d_in[8];
  const int*   ix1 = (const int*)  d_in[9];
  const int*   iy1 = (const int*)  d_in[10];
  const float* w1  = (const float*)d_in[11];

  float* out   = (float*)d_out;
  float* parts = (float*)d_ws;   // 64 floats, fully rewritten every call

  envmap_render_main<<<NPIX / 256, 256, 0, stream>>>(
      x, y, bs0, bd0, ix0, iy0, w0, bs1, bd1, ix1, iy1, w1, out, parts);
  envmap_finalize_loss<<<1, 32, 0, stream>>>(parts, out);
}